// Loss_ssi_83227876262122
// MI455X (gfx1250) — compile-verified
//
#include <hip/hip_runtime.h>
#include <hip/hip_bf16.h>

// SSI (scale/shift-invariant) loss for MI455X (gfx1250, wave32).
//
// Per (B,N) image: exact masked median via 3-level radix-select on the
// order-preserving float key (LDS histograms, integer ds atomics), masked MAD,
// then per-H-row sum(rho)/max(cnt,1) accumulated with exact f32 WMMA
// (V_WMMA_F32_16X16X4_F32, B = ones). One workgroup per image; deterministic.
//
// v3: alignment-specialized inner loops hoisted out of the hot path,
//     __builtin_assume_aligned so float2/float4 loads lower to b64/b128.

#define SSI_EPS 1e-8f

#if defined(__has_builtin)
#if __has_builtin(__builtin_amdgcn_wmma_f32_16x16x4_f32)
#define SSI_HAVE_WMMA_F32X4 1
#endif
#endif

typedef __attribute__((ext_vector_type(2))) float v2f;
typedef __attribute__((ext_vector_type(8))) float v8f;

// Monotonic unsigned key for float ordering (handles negatives / inf).
__device__ __forceinline__ unsigned ssiKey(float f) {
    unsigned u = __float_as_uint(f);
    return u ^ ((u >> 31) ? 0xFFFFFFFFu : 0x80000000u);
}
__device__ __forceinline__ float ssiKeyInv(unsigned k) {
    unsigned u = k ^ ((k & 0x80000000u) ? 0x80000000u : 0xFFFFFFFFu);
    return __uint_as_float(u);
}

// Serial histogram selection (run by one thread; ~2K LDS reads, negligible).
__device__ __forceinline__ unsigned ssiSelectBin(const unsigned* hist, int bins,
                                                 unsigned& rank) {
    unsigned cum = 0;
    for (int i = 0; i < bins; ++i) {
        unsigned h = hist[i];
        if (rank < cum + h) { rank -= cum; return (unsigned)i; }
        cum += h;
    }
    rank = 0;
    return (unsigned)(bins - 1);
}

// Deterministic block reduction (wave32 shuffle + LDS). Result valid on tid 0.
__device__ __forceinline__ float ssiBlockReduce(float v, float* scratch) {
    for (int off = 16; off > 0; off >>= 1) v += __shfl_down(v, off, 32);
    const int lane = threadIdx.x & 31;
    const int wid  = threadIdx.x >> 5;
    if (lane == 0) scratch[wid] = v;
    __syncthreads();
    const int nw = (blockDim.x + 31) >> 5;
    float r = 0.0f;
    if (wid == 0) {
        r = (lane < nw) ? scratch[lane] : 0.0f;
        for (int off = 16; off > 0; off >>= 1) r += __shfl_down(r, off, 32);
    }
    __syncthreads();
    return r;
}

// Stream (p, q, mask) over one image; f(p, q, mask_bool) per element.
// Vector path: float4 / b128 loads + uint mask word (requires HW % 4 == 0 so
// every row base stays 16B aligned). Loads are unconditional: masked-out
// elements share cachelines, so this costs no extra HBM traffic.
template <typename F>
__device__ __forceinline__ void ssiForEach(const float* __restrict__ Pb,
                                           const float* __restrict__ Qb,
                                           const unsigned char* __restrict__ Mb,
                                           int HW, int tid, int nthr, bool vecOK,
                                           F f) {
    if (vecOK) {
        const int HW4 = HW >> 2;
        const int pfStep = nthr << 5;  // prefetch ~8 vector iters ahead
        for (int g = tid; g < HW4; g += nthr) {
            const int i = g << 2;
            if (i + pfStep < HW) {
                __builtin_prefetch(Pb + i + pfStep, 0, 0);
                __builtin_prefetch(Qb + i + pfStep, 0, 0);
            }
            const float4 p =
                *(const float4*)__builtin_assume_aligned(Pb + i, 16);
            const float4 q =
                *(const float4*)__builtin_assume_aligned(Qb + i, 16);
            const unsigned mw =
                *(const unsigned*)__builtin_assume_aligned(Mb + i, 4);
            f(p.x, q.x, (mw & 0x000000FFu) != 0u);
            f(p.y, q.y, (mw & 0x0000FF00u) != 0u);
            f(p.z, q.z, (mw & 0x00FF0000u) != 0u);
            f(p.w, q.w, (mw & 0xFF000000u) != 0u);
        }
    } else {
        for (int i = tid; i < HW; i += nthr) f(Pb[i], Qb[i], Mb[i] != 0);
    }
}

__global__ void __launch_bounds__(512)
ssi_row_kernel(const float* __restrict__ P, const float* __restrict__ Q,
               const unsigned char* __restrict__ Mk,
               float* __restrict__ partial, int H, int W) {
    const int HW = H * W;
    const size_t base = (size_t)blockIdx.x * (size_t)HW;
    const float* __restrict__ Pb = P + base;
    const float* __restrict__ Qb = Q + base;
    const unsigned char* __restrict__ Mb = Mk + base;
    const int tid  = threadIdx.x;
    const int nthr = blockDim.x;
    const bool vecOK = ((HW & 3) == 0);

    __shared__ unsigned histP[2048];
    __shared__ unsigned histQ[2048];
    __shared__ float    red[32];
    __shared__ unsigned cntS;
    __shared__ unsigned selP, selQ, rankP, rankQ;
    __shared__ float    medPs, medQs, iscPs, iscQs;

    // ---------- Pass 1: count + top-11-bit histograms ----------
    for (int i = tid; i < 2048; i += nthr) { histP[i] = 0u; histQ[i] = 0u; }
    if (tid == 0) cntS = 0u;
    __syncthreads();

    unsigned lc = 0;
    ssiForEach(Pb, Qb, Mb, HW, tid, nthr, vecOK,
               [&](float p, float q, bool m) {
                   if (m) {
                       ++lc;
                       atomicAdd(&histP[ssiKey(p) >> 21], 1u);
                       atomicAdd(&histQ[ssiKey(q) >> 21], 1u);
                   }
               });
    atomicAdd(&cntS, lc);
    __syncthreads();

    if (tid == 0) {
        const unsigned cnt = cntS;
        const unsigned k = cnt ? ((cnt - 1u) >> 1) : 0u;  // lower-middle median
        unsigned r = k; selP = ssiSelectBin(histP, 2048, r); rankP = r;
        r = k;          selQ = ssiSelectBin(histQ, 2048, r); rankQ = r;
    }
    __syncthreads();
    for (int i = tid; i < 2048; i += nthr) { histP[i] = 0u; histQ[i] = 0u; }
    __syncthreads();

    // ---------- Pass 2: middle 11 bits ----------
    {
        const unsigned sp = selP, sq = selQ;
        ssiForEach(Pb, Qb, Mb, HW, tid, nthr, vecOK,
                   [&](float p, float q, bool m) {
                       if (m) {
                           const unsigned kp = ssiKey(p);
                           if ((kp >> 21) == sp)
                               atomicAdd(&histP[(kp >> 10) & 0x7FFu], 1u);
                           const unsigned kq = ssiKey(q);
                           if ((kq >> 21) == sq)
                               atomicAdd(&histQ[(kq >> 10) & 0x7FFu], 1u);
                       }
                   });
    }
    __syncthreads();
    if (tid == 0) {
        unsigned r = rankP, b = ssiSelectBin(histP, 2048, r);
        rankP = r; selP = (selP << 11) | b;
        r = rankQ; b = ssiSelectBin(histQ, 2048, r);
        rankQ = r; selQ = (selQ << 11) | b;
    }
    __syncthreads();
    for (int i = tid; i < 2048; i += nthr) { histP[i] = 0u; histQ[i] = 0u; }
    __syncthreads();

    // ---------- Pass 3: low 10 bits -> exact median ----------
    {
        const unsigned sp = selP, sq = selQ;
        ssiForEach(Pb, Qb, Mb, HW, tid, nthr, vecOK,
                   [&](float p, float q, bool m) {
                       if (m) {
                           const unsigned kp = ssiKey(p);
                           if ((kp >> 10) == sp)
                               atomicAdd(&histP[kp & 0x3FFu], 1u);
                           const unsigned kq = ssiKey(q);
                           if ((kq >> 10) == sq)
                               atomicAdd(&histQ[kq & 0x3FFu], 1u);
                       }
                   });
    }
    __syncthreads();
    if (tid == 0) {
        const unsigned cnt = cntS;
        unsigned r = rankP, b = ssiSelectBin(histP, 1024, r);
        medPs = cnt ? ssiKeyInv((selP << 10) | b) : 0.0f;
        r = rankQ; b = ssiSelectBin(histQ, 1024, r);
        medQs = cnt ? ssiKeyInv((selQ << 10) | b) : 0.0f;
    }
    __syncthreads();

    // ---------- Pass 4: MAD (branchless: mask as multiplier) ----------
    const float medP = medPs, medQ = medQs;
    float sP = 0.0f, sQ = 0.0f;
    ssiForEach(Pb, Qb, Mb, HW, tid, nthr, vecOK,
               [&](float p, float q, bool m) {
                   const float mf = m ? 1.0f : 0.0f;
                   sP += mf * fabsf(p - medP);
                   sQ += mf * fabsf(q - medQ);
               });
    sP = ssiBlockReduce(sP, red);
    sQ = ssiBlockReduce(sQ, red);
    if (tid == 0) {
        const float cf = fmaxf((float)cntS, 1.0f);
        const float scP = (cntS > 0u) ? (sP / cf + SSI_EPS) : SSI_EPS;
        const float scQ = (cntS > 0u) ? (sQ / cf + SSI_EPS) : SSI_EPS;
        iscPs = 1.0f / scP;
        iscQs = 1.0f / scQ;
    }
    __syncthreads();
    const float iscP = iscPs, iscQ = iscQs;

    // ---------- Pass 5: loss = sum_h [ sum_w rho / max(cnt_h,1) ] ----------
    float localLoss = 0.0f;
    const int lane   = tid & 31;
    const int wave   = tid >> 5;
    const int nwaves = nthr >> 5;

#ifdef SSI_HAVE_WMMA_F32X4
    // Exact f32 WMMA row sums: D(16x16) = A(16x4) * ones(4x16) + C.
    // A layout (f32 16x4): lanes 0-15 hold M=lane, K={0,1} in VGPR0/1;
    // lanes 16-31 hold M=lane-16, K={2,3}. B all-ones => every D column is
    // the K-reduction; column N=0 sits in lanes 0 (M=0..7) / 16 (M=8..15).
    // Branchless bodies; alignment specialization hoisted out of the loop.
    {
        const int HT    = (H + 15) >> 4;   // 16-row tiles (last padded)
        const int m     = lane & 15;
        const int klo   = (lane < 16) ? 0 : 2;
        const int Wfull = W & ~3;          // full 4-wide chunks
        const bool w2ok = ((W & 1) == 0);  // even W => ro and ro+w even
        const v2f bOnes = {1.0f, 1.0f};

        for (int t = wave; t < HT; t += nwaves) {
            const int   h      = (t << 4) + m;
            const float hvalid = (h < H) ? 1.0f : 0.0f;
            const int   hc     = (h < H) ? h : (H - 1);
            const size_t ro    = (size_t)hc * (size_t)W;
            v8f accS = {0.f, 0.f, 0.f, 0.f, 0.f, 0.f, 0.f, 0.f};
            v8f accC = {0.f, 0.f, 0.f, 0.f, 0.f, 0.f, 0.f, 0.f};

            // Accumulate one 16x4 chunk pair into accS/accC.
            auto wmmaAcc = [&](float p0, float p1, float q0, float q1,
                               float f0, float f1) {
                const float d0 = (p0 - medP) * iscP - (q0 - medQ) * iscQ;
                const float d1 = (p1 - medP) * iscP - (q1 - medQ) * iscQ;
                const v2f aS = {f0 * d0 * d0, f1 * d1 * d1};
                const v2f aC = {f0, f1};
                // (neg_a, A, neg_b, B, c_mod, C, reuse_a, reuse_b)
                accS = __builtin_amdgcn_wmma_f32_16x16x4_f32(
                    false, aS, false, bOnes, (short)0, accS, false, false);
                accC = __builtin_amdgcn_wmma_f32_16x16x4_f32(
                    false, aC, false, bOnes, (short)0, accC, false, false);
            };

            if (w2ok) {
                // Aligned pair loads -> global_load_b64 / b16.
                for (int w0 = 0; w0 < Wfull; w0 += 4) {
                    const int w = w0 + klo;  // even; in-bounds incl. w+1
                    const float2 pv = *(const float2*)__builtin_assume_aligned(
                        Pb + ro + w, 8);
                    const float2 qv = *(const float2*)__builtin_assume_aligned(
                        Qb + ro + w, 8);
                    const unsigned short ms =
                        *(const unsigned short*)__builtin_assume_aligned(
                            Mb + ro + w, 2);
                    const float f0 = (ms & 0x00FFu) ? hvalid : 0.0f;
                    const float f1 = (ms & 0xFF00u) ? hvalid : 0.0f;
                    wmmaAcc(pv.x, pv.y, qv.x, qv.y, f0, f1);
                }
            } else {
                for (int w0 = 0; w0 < Wfull; w0 += 4) {
                    const int w = w0 + klo;
                    const float f0 = Mb[ro + w]     ? hvalid : 0.0f;
                    const float f1 = Mb[ro + w + 1] ? hvalid : 0.0f;
                    wmmaAcc(Pb[ro + w], Pb[ro + w + 1],
                            Qb[ro + w], Qb[ro + w + 1], f0, f1);
                }
            }

            if (Wfull < W) {               // single tail chunk, clamped loads
                const int w   = Wfull + klo;
                const int wc0 = (w < W) ? w : (W - 1);
                const int wc1 = (w + 1 < W) ? (w + 1) : (W - 1);
                const float v0 = (w < W) ? hvalid : 0.0f;
                const float v1 = (w + 1 < W) ? hvalid : 0.0f;
                const float f0 = Mb[ro + wc0] ? v0 : 0.0f;
                const float f1 = Mb[ro + wc1] ? v1 : 0.0f;
                wmmaAcc(Pb[ro + wc0], Pb[ro + wc1],
                        Qb[ro + wc0], Qb[ro + wc1], f0, f1);
            }

            if (lane == 0 || lane == 16) {
#pragma unroll
                for (int i2 = 0; i2 < 8; ++i2)
                    localLoss += accS[i2] / fmaxf(accC[i2], 1.0f);
            }
        }
    }
#else
    // Fallback: wave-per-h-row shuffle reduction (still exact f32).
    for (int h = wave; h < H; h += nwaves) {
        const size_t ro = (size_t)h * (size_t)W;
        float s = 0.0f, c = 0.0f;
        for (int w = lane; w < W; w += 32) {
            const float mf = Mb[ro + w] ? 1.0f : 0.0f;
            const float dp = (Pb[ro + w] - medP) * iscP;
            const float dq = (Qb[ro + w] - medQ) * iscQ;
            const float d  = dp - dq;
            s += mf * d * d; c += mf;
        }
        for (int off = 16; off > 0; off >>= 1) {
            s += __shfl_down(s, off, 32);
            c += __shfl_down(c, off, 32);
        }
        if (lane == 0) localLoss += s / fmaxf(c, 1.0f);
    }
#endif

    const float tot = ssiBlockReduce(localLoss, red);
    if (tid == 0) partial[blockIdx.x] = tot;
}

__global__ void ssi_finalize_kernel(const float* __restrict__ partial,
                                    float* __restrict__ out, int BN,
                                    float invDen) {
    __shared__ float red[32];
    float v = 0.0f;
    for (int i = threadIdx.x; i < BN; i += blockDim.x) v += partial[i];
    v = ssiBlockReduce(v, red);
    if (threadIdx.x == 0) out[0] = v * invDen;
}

extern "C" void kernel_launch(void* const* d_in, const int* in_sizes, int n_in,
                              void* d_out, int out_size, void* d_ws, size_t ws_size,
                              hipStream_t stream) {
    (void)n_in; (void)out_size; (void)ws_size;
    const float* P = (const float*)d_in[0];                 // pred  f32 [B,N,H,W]
    const float* Q = (const float*)d_in[1];                 // y     f32 [B,N,H,W]
    const unsigned char* M = (const unsigned char*)d_in[2]; // bool mask, 1B/elem

    const int H = 518, W = 518;
    const int HW = H * W;
    const int BN = in_sizes[0] / HW;                        // 128 for the reference

    float* partial = (float*)d_ws;                          // BN floats of scratch

    ssi_row_kernel<<<dim3(BN), dim3(512), 0, stream>>>(P, Q, M, partial, H, W);
    const float invDen = 1.0f / ((float)BN * (float)H);
    ssi_finalize_kernel<<<dim3(1), dim3(128), 0, stream>>>(partial, (float*)d_out,
                                                           BN, invDen);
}